// S4Decoder_63900523430517
// MI455X (gfx1250) — compile-verified
//
#include <hip/hip_runtime.h>
#include <hip/hip_bf16.h>
#include <math.h>

typedef __attribute__((ext_vector_type(2))) float v2f;
typedef __attribute__((ext_vector_type(8))) float v8f;

#define BB 8
#define HH 64
#define LIN 1024
#define KCONV 16
#define LL (LIN*KCONV)      // 16384
#define NN 32               // complex modes
#define TCH 256             // scan chunk length
#define CCH (LL/TCH)        // 64 chunks
#define NBLK 4

__device__ __forceinline__ float gelu_f(float x){
    float x3 = x*x*x;
    return 0.5f*x*(1.0f + tanhf(0.7978845608028654f*(x + 0.044715f*x3)));
}
__device__ __forceinline__ float sigm_f(float x){ return 1.0f/(1.0f + expf(-x)); }
__device__ __forceinline__ void lds_fence(){ asm volatile("s_wait_dscnt 0" ::: "memory"); }

// ---- ConvTranspose1d(1->H, k=16, stride=16): u[b,h,lin*16+k] = x[b,lin]*w[h,k]+b[h]
__global__ void conv_up_kernel(const float* __restrict__ x, const float* __restrict__ cw,
                               const float* __restrict__ cb, float* __restrict__ u){
    size_t idx = (size_t)blockIdx.x*blockDim.x + threadIdx.x;   // B*H*L
    int l = (int)(idx % LL); int bh = (int)(idx / LL); int h = bh % HH; int b = bh / HH;
    if (b >= BB) return;
    u[idx] = fmaf(x[b*LIN + (l>>4)], cw[h*KCONV + (l&15)], cb[h]);
}

// ---- LayerNorm over channel dim (stride-L layout), one thread per (b,l)
__global__ void ln_ch_kernel(const float* __restrict__ u, const float* __restrict__ w,
                             const float* __restrict__ bias, float* __restrict__ z){
    int idx = blockIdx.x*blockDim.x + threadIdx.x;   // B*L
    int l = idx % LL; int b = idx / LL;
    if (b >= BB) return;
    const float* up = u + (size_t)b*HH*LL + l;
    float s = 0.f, s2 = 0.f;
    #pragma unroll 8
    for (int h = 0; h < HH; ++h){ float v = up[(size_t)h*LL]; s += v; s2 += v*v; }
    float mu = s * (1.0f/HH);
    float var = s2*(1.0f/HH) - mu*mu;
    float rs = rsqrtf(var + 1e-5f);
    float* zp = z + (size_t)b*HH*LL + l;
    #pragma unroll 8
    for (int h = 0; h < HH; ++h){
        float v = up[(size_t)h*LL];
        zp[(size_t)h*LL] = fmaf((v - mu)*rs, w[h], bias[h]);
    }
}

// ---- S4D discretization: w = exp(dt*A), Ct = 2*C*(w-1)/A, wT = w^TCH
__global__ void s4pre_kernel(const float* __restrict__ log_dt, const float* __restrict__ A_re,
                             const float* __restrict__ A_im, const float* __restrict__ C_re,
                             const float* __restrict__ C_im, float2* __restrict__ wC,
                             float2* __restrict__ ctC, float2* __restrict__ wTC){
    int idx = blockIdx.x*blockDim.x + threadIdx.x;  // H*N
    if (idx >= HH*NN) return;
    int h = idx / NN;
    float dt = expf(log_dt[h]);
    float ar = A_re[idx], ai = A_im[idx];
    float dr = ar*dt, di = ai*dt;
    float er = expf(dr);
    float wr = er*cosf(di), wi = er*sinf(di);
    float mr = wr - 1.0f, mi = wi;                       // w-1
    float den = 1.0f/(ar*ar + ai*ai);
    float qr = (mr*ar + mi*ai)*den;                      // (w-1)/A
    float qi = (mi*ar - mr*ai)*den;
    float cr = C_re[idx], ci = C_im[idx];
    wC[idx]  = make_float2(wr, wi);
    ctC[idx] = make_float2(2.0f*(cr*qr - ci*qi), 2.0f*(cr*qi + ci*qr));
    float eT = expf(dr*(float)TCH);
    wTC[idx] = make_float2(eT*cosf(di*(float)TCH), eT*sinf(di*(float)TCH));
}

// ---- Scan pass A: per-(b,h,chunk) local end-state, lane = mode
__global__ void s4_local_kernel(const float* __restrict__ z, const float2* __restrict__ wC,
                                float2* __restrict__ states){
    int widx = blockIdx.x*blockDim.y + threadIdx.y;   // B*H*C waves
    int lane = threadIdx.x;
    int c = widx % CCH; int hh = (widx / CCH) % HH; int b = widx / (CCH*HH);
    if (b >= BB) return;
    float2 w = wC[hh*NN + lane];
    const float* zp = z + ((size_t)(b*HH + hh))*LL + c*TCH;
    float sr = 0.f, si = 0.f;
    for (int g = 0; g < TCH/32; ++g){
        float zv = zp[g*32 + lane];
        #pragma unroll
        for (int j = 0; j < 32; ++j){
            float zj = __shfl(zv, j, 32);
            float nr = fmaf(w.x, sr, fmaf(-w.y, si, zj));
            float ni = fmaf(w.x, si, w.y*sr);
            sr = nr; si = ni;
        }
    }
    states[((size_t)(b*HH + hh)*CCH + c)*NN + lane] = make_float2(sr, si);
}

// ---- Scan pass B: exclusive carry scan over chunks, thread per (b,h,n)
__global__ void s4_carry_kernel(const float2* __restrict__ wTC, float2* __restrict__ states){
    int idx = blockIdx.x*blockDim.x + threadIdx.x;    // B*H*N
    if (idx >= BB*HH*NN) return;
    int n = idx % NN; int hh = (idx / NN) % HH; int b = idx / (NN*HH);
    float2 wT = wTC[hh*NN + n];
    float cr = 0.f, ci = 0.f;
    float2* sp = states + ((size_t)(b*HH + hh))*CCH*NN + n;
    for (int c = 0; c < CCH; ++c){
        float2 loc = sp[(size_t)c*NN];
        sp[(size_t)c*NN] = make_float2(cr, ci);
        float nr = fmaf(wT.x, cr, -wT.y*ci) + loc.x;
        float ni = fmaf(wT.x, ci,  wT.y*cr) + loc.y;
        cr = nr; ci = ni;
    }
}

// ---- Scan pass C: replay with carry, emit gelu(y + D*z). LDS 32x33 transpose
// amortizes the cross-mode reduction (conflict-free on 64 banks).
__global__ void s4_apply_kernel(const float* __restrict__ z, const float2* __restrict__ wC,
                                const float2* __restrict__ ctC, const float2* __restrict__ states,
                                const float* __restrict__ Dvec, float* __restrict__ out){
    __shared__ float tile[4][32*33];
    int wy = threadIdx.y;
    int widx = blockIdx.x*blockDim.y + wy;
    int lane = threadIdx.x;
    int c = widx % CCH; int hh = (widx / CCH) % HH; int b = widx / (CCH*HH);
    if (b >= BB) return;
    float2 w  = wC[hh*NN + lane];
    float2 ct = ctC[hh*NN + lane];
    float2 s0 = states[((size_t)(b*HH + hh)*CCH + c)*NN + lane];
    float sr = s0.x, si = s0.y;
    float Dh = Dvec[hh];
    size_t base = ((size_t)(b*HH + hh))*LL + c*TCH;
    const float* zp = z + base;
    float* op = out + base;
    float p[32];
    for (int g = 0; g < TCH/32; ++g){
        float zv = zp[g*32 + lane];
        #pragma unroll
        for (int j = 0; j < 32; ++j){
            float zj = __shfl(zv, j, 32);
            float nr = fmaf(w.x, sr, fmaf(-w.y, si, zj));
            float ni = fmaf(w.x, si, w.y*sr);
            sr = nr; si = ni;
            p[j] = fmaf(-ct.y, si, ct.x*sr);
        }
        #pragma unroll
        for (int j = 0; j < 32; ++j) tile[wy][lane*33 + j] = p[j];
        lds_fence();
        float acc = 0.f;
        #pragma unroll
        for (int j = 0; j < 32; ++j) acc += tile[wy][j*33 + lane];
        lds_fence();
        op[g*32 + lane] = gelu_f(fmaf(Dh, zv, acc));
    }
}

// ---- outlin (128x64) GEMM fused with GLU residual: u += a*sigmoid(g)
__global__ void outlin_glu_kernel(const float* __restrict__ W, const float* __restrict__ bias,
                                  const float* __restrict__ X, float* __restrict__ U){
    int lane = threadIdx.x;
    int n0 = blockIdx.x*16;
    int m0 = (int)threadIdx.y*16;           // 0..48 over 64 'a' rows
    int b = blockIdx.z;
    int col = lane & 15, half = lane >> 4;
    int krow = half*2;
    v8f ca = {}; v8f cg = {};
    const float* Wa = W + (size_t)(m0 + col)*HH;
    const float* Wg = W + (size_t)(m0 + 64 + col)*HH;
    const float* Xp = X + (size_t)b*HH*LL + n0 + col;
    #pragma unroll
    for (int k = 0; k < HH; k += 4){
        v2f a, gw, bb;
        a.x  = Wa[k + krow];  a.y  = Wa[k + krow + 1];
        gw.x = Wg[k + krow];  gw.y = Wg[k + krow + 1];
        bb.x = Xp[(size_t)(k + krow)*LL];
        bb.y = Xp[(size_t)(k + krow + 1)*LL];
        ca = __builtin_amdgcn_wmma_f32_16x16x4_f32(false, a,  false, bb, (short)0, ca, false, false);
        cg = __builtin_amdgcn_wmma_f32_16x16x4_f32(false, gw, false, bb, (short)0, cg, false, false);
    }
    float* Up = U + (size_t)b*HH*LL;
    #pragma unroll
    for (int r = 0; r < 8; ++r){
        int row = m0 + r + 8*half;
        size_t oi = (size_t)row*LL + n0 + col;
        float av = ca[r] + bias[row];
        float gv = cg[r] + bias[64 + row];
        Up[oi] += av * sigm_f(gv);
    }
}

// ---- Generic channel GEMM: Y[b,m,l] = act(sum_k W[m,k]*X[b,k,l] + bias[m]) (+Y)
__global__ void gemm_kernel(const float* __restrict__ W, const float* __restrict__ bias,
                            const float* __restrict__ X, float* __restrict__ Y,
                            int M, int K, int act, int residual){
    int lane = threadIdx.x;
    int n0 = blockIdx.x*16;
    int m0 = (blockIdx.y*blockDim.y + threadIdx.y)*16;
    int b = blockIdx.z;
    int col = lane & 15, half = lane >> 4;
    int krow = half*2;
    v8f c = {};
    const float* Wp = W + (size_t)(m0 + col)*K;
    const float* Xp = X + (size_t)b*K*LL + n0 + col;
    for (int k = 0; k < K; k += 4){
        v2f a, bb;
        a.x = Wp[k + krow]; a.y = Wp[k + krow + 1];
        bb.x = Xp[(size_t)(k + krow)*LL];
        bb.y = Xp[(size_t)(k + krow + 1)*LL];
        c = __builtin_amdgcn_wmma_f32_16x16x4_f32(false, a, false, bb, (short)0, c, false, false);
    }
    float* Yp = Y + (size_t)b*M*LL;
    #pragma unroll
    for (int r = 0; r < 8; ++r){
        int row = m0 + r + 8*half;
        size_t oi = (size_t)row*LL + n0 + col;
        float v = c[r] + bias[row];
        if (act) v = gelu_f(v);
        if (residual) v += Yp[oi];
        Yp[oi] = v;
    }
}

// ---- final projection H->1
__global__ void proj_kernel(const float* __restrict__ U, const float* __restrict__ pw,
                            const float* __restrict__ pb, float* __restrict__ out){
    int idx = blockIdx.x*blockDim.x + threadIdx.x;   // B*L
    int l = idx % LL; int b = idx / LL;
    if (b >= BB) return;
    const float* up = U + (size_t)b*HH*LL + l;
    float acc = pb[0];
    #pragma unroll 8
    for (int h = 0; h < HH; ++h) acc = fmaf(up[(size_t)h*LL], pw[h], acc);
    out[idx] = acc;
}

extern "C" void kernel_launch(void* const* d_in, const int* in_sizes, int n_in,
                              void* d_out, int out_size, void* d_ws, size_t ws_size,
                              hipStream_t stream){
    (void)in_sizes; (void)n_in; (void)out_size; (void)ws_size;
    const float* x       = (const float*)d_in[0];
    const float* conv_w  = (const float*)d_in[1];
    const float* conv_b  = (const float*)d_in[2];
    const float* ln1_w   = (const float*)d_in[3];
    const float* ln1_b   = (const float*)d_in[4];
    const float* log_dt  = (const float*)d_in[5];
    const float* A_re    = (const float*)d_in[6];
    const float* A_im    = (const float*)d_in[7];
    const float* C_re    = (const float*)d_in[8];
    const float* C_im    = (const float*)d_in[9];
    const float* Dv      = (const float*)d_in[10];
    const float* outlin_w= (const float*)d_in[11];
    const float* outlin_b= (const float*)d_in[12];
    const float* ln2_w   = (const float*)d_in[13];
    const float* ln2_b   = (const float*)d_in[14];
    const float* ff_w1   = (const float*)d_in[15];
    const float* ff_b1   = (const float*)d_in[16];
    const float* ff_w2   = (const float*)d_in[17];
    const float* ff_b2   = (const float*)d_in[18];
    const float* proj_w  = (const float*)d_in[19];
    const float* proj_b  = (const float*)d_in[20];
    float* out = (float*)d_out;

    const size_t nBHL = (size_t)BB*HH*LL;          // 8.39M floats
    float*  u      = (float*)d_ws;
    float*  z      = u + nBHL;
    float*  yc     = z + nBHL;
    float*  gl     = yc + nBHL;                    // (B,128,L)
    float2* states = (float2*)(gl + 2*nBHL);
    float2* wC     = states + (size_t)BB*HH*CCH*NN;
    float2* ctC    = wC + HH*NN;
    float2* wTC    = ctC + HH*NN;

    dim3 b256(256);
    dim3 wblk(32, 4);
    conv_up_kernel<<<(unsigned)(nBHL/256), b256, 0, stream>>>(x, conv_w, conv_b, u);

    const int nw = BB*HH*CCH;   // scan waves
    for (int i = 0; i < NBLK; ++i){
        ln_ch_kernel<<<(BB*LL)/256, b256, 0, stream>>>(u, ln1_w + i*HH, ln1_b + i*HH, z);
        s4pre_kernel<<<(HH*NN)/256, b256, 0, stream>>>(log_dt + i*HH, A_re + i*HH*NN, A_im + i*HH*NN,
                                                       C_re + i*HH*NN, C_im + i*HH*NN, wC, ctC, wTC);
        s4_local_kernel<<<nw/4, wblk, 0, stream>>>(z, wC, states);
        s4_carry_kernel<<<(BB*HH*NN)/256, b256, 0, stream>>>(wTC, states);
        s4_apply_kernel<<<nw/4, wblk, 0, stream>>>(z, wC, ctC, states, Dv + i*HH, yc);
        outlin_glu_kernel<<<dim3(LL/16, 1, BB), wblk, 0, stream>>>(outlin_w + i*2*HH*HH,
                                                                   outlin_b + i*2*HH, yc, u);
        ln_ch_kernel<<<(BB*LL)/256, b256, 0, stream>>>(u, ln2_w + i*HH, ln2_b + i*HH, z);
        gemm_kernel<<<dim3(LL/16, 2, BB), wblk, 0, stream>>>(ff_w1 + i*2*HH*HH, ff_b1 + i*2*HH,
                                                             z, gl, 2*HH, HH, 1, 0);
        gemm_kernel<<<dim3(LL/16, 1, BB), wblk, 0, stream>>>(ff_w2 + i*2*HH*HH, ff_b2 + i*HH,
                                                             gl, u, HH, 2*HH, 0, 1);
    }
    proj_kernel<<<(BB*LL)/256, b256, 0, stream>>>(u, proj_w, proj_b, out);
}